// GINELayer_2954937499916
// MI455X (gfx1250) — compile-verified
//
#include <hip/hip_runtime.h>
#include <hip/hip_bf16.h>

#define N_NODES 100000
#define N_EDGES 600000
#define DIM 128
#define EDGE_DIM 32
#define LDS_STRIDE (DIM + 4)   // pad to break 64-bank conflicts on row-wise reads

typedef float v2f __attribute__((ext_vector_type(2)));
typedef float v8f __attribute__((ext_vector_type(8)));

// ---------------------------------------------------------------------------
// Zero the aggregation workspace (d_ws is poisoned, must re-zero each launch)
// ---------------------------------------------------------------------------
__global__ __launch_bounds__(256) void zero_agg_kernel(float4* __restrict__ p, int n4) {
    int idx = blockIdx.x * blockDim.x + threadIdx.x;
    if (idx < n4) p[idx] = make_float4(0.f, 0.f, 0.f, 0.f);
}

// ---------------------------------------------------------------------------
// Edge kernel: msg = edge_attr @ We + be + x[j], scatter-add into agg[i]
// One wave32 handles a tile of 16 edges, full 128 output dims.
// GEMM done with V_WMMA_F32_16X16X4_F32 (exact fp32), 8 K-steps x 8 N-tiles.
// ---------------------------------------------------------------------------
__global__ __launch_bounds__(128) void edge_kernel(
    const float* __restrict__ x,
    const int*   __restrict__ ei,     // [2, E]: row0 = i (dst), row1 = j (src)
    const float* __restrict__ ea,     // [E, 32]
    const float* __restrict__ We,     // [32, 128]
    const float* __restrict__ be,     // [128]
    float*       __restrict__ agg)    // [N, 128]
{
    const int lane = threadIdx.x & 31;
    const int wave = threadIdx.x >> 5;
    const int tile = blockIdx.x * 4 + wave;   // 37500 tiles exactly
    const int base = tile * 16;
    const int m    = lane & 15;               // A layout: lane -> M row
    const int hi   = lane >> 4;               // hi half holds K+2

    // A fragments: edge_attr[base+m][k], k = 4*s + 2*hi + {0,1}
    v2f a[8];
    const float* ar = ea + (size_t)(base + m) * EDGE_DIM + 2 * hi;
    #pragma unroll
    for (int s = 0; s < 8; ++s) {
        a[s].x = ar[4 * s];
        a[s].y = ar[4 * s + 1];
    }

    // Edge endpoints for the rows this lane emits (row = r + 8*hi).
    // Hoist row base pointers out of the nt loop: address math done once,
    // and unsigned casts avoid per-access sign-extension.
    const int* dsti = ei;
    const int* srcj = ei + N_EDGES;
    const float* xrow[8];
    float*       grow[8];
    #pragma unroll
    for (int r = 0; r < 8; ++r) {
        int e = base + r + 8 * hi;
        xrow[r] = x   + (size_t)(unsigned)srcj[e] * DIM;
        grow[r] = agg + (size_t)(unsigned)dsti[e] * DIM;
    }

    #pragma unroll
    for (int nt = 0; nt < 8; ++nt) {
        const int col = nt * 16 + m;          // C/D layout: lane&15 -> N col
        v8f c = {};
        #pragma unroll
        for (int s = 0; s < 8; ++s) {
            const int kb = 4 * s + 2 * hi;    // B layout mirrors A with N in lanes
            v2f b;
            b.x = We[(size_t)kb * DIM + col];
            b.y = We[(size_t)(kb + 1) * DIM + col];
            c = __builtin_amdgcn_wmma_f32_16x16x4_f32(
                    false, a[s], false, b, (short)0, c, false, false);
        }
        const float bias = be[col];
        #pragma unroll
        for (int r = 0; r < 8; ++r) {
            // C/D layout: VGPR r holds row (r + 8*hi)
            float v = c[r] + bias + xrow[r][col];
            unsafeAtomicAdd(&grow[r][col], v);   // global_atomic_add_f32
        }
    }
}

// ---------------------------------------------------------------------------
// Node kernel: h = (1+eps)*x + agg; out = LN(relu(h@W1+b1)@W2 + b2)
// One wave32 per 16 nodes. Two fp32 WMMA GEMMs with an LDS relayout between.
// ---------------------------------------------------------------------------
__global__ __launch_bounds__(32) void node_kernel(
    const float* __restrict__ x,
    const float* __restrict__ agg,
    const float* __restrict__ W1, const float* __restrict__ b1,
    const float* __restrict__ W2, const float* __restrict__ b2,
    const float* __restrict__ epsp,
    const float* __restrict__ gamma, const float* __restrict__ beta,
    float* __restrict__ out)
{
    __shared__ float t_s[16][LDS_STRIDE];   // relu(h@W1+b1)
    __shared__ float o_s[16][LDS_STRIDE];   // t@W2+b2
    __shared__ float mu_s[16], rs_s[16];

    const int lane = threadIdx.x;
    const int m    = lane & 15;
    const int hi   = lane >> 4;
    const int base = blockIdx.x * 16;       // 6250 blocks exactly
    const float scale = 1.0f + epsp[0];

    // ---- build h A-fragments: k = 4*s + 2*hi + {0,1}, 32 K-steps ----
    v2f a[32];
    {
        const float* xr = x   + (size_t)(base + m) * DIM + 2 * hi;
        const float* gr = agg + (size_t)(base + m) * DIM + 2 * hi;
        #pragma unroll
        for (int s = 0; s < 32; ++s) {
            a[s].x = scale * xr[4 * s]     + gr[4 * s];
            a[s].y = scale * xr[4 * s + 1] + gr[4 * s + 1];
        }
    }

    // ---- GEMM1: t = relu(h @ W1 + b1) ----
    #pragma unroll
    for (int nt = 0; nt < 8; ++nt) {
        const int col = nt * 16 + m;
        v8f c = {};
        #pragma unroll
        for (int s = 0; s < 32; ++s) {
            const int kb = 4 * s + 2 * hi;
            v2f b;
            b.x = W1[(size_t)kb * DIM + col];
            b.y = W1[(size_t)(kb + 1) * DIM + col];
            c = __builtin_amdgcn_wmma_f32_16x16x4_f32(
                    false, a[s], false, b, (short)0, c, false, false);
        }
        const float bias = b1[col];
        #pragma unroll
        for (int r = 0; r < 8; ++r) {
            float v = c[r] + bias;
            t_s[r + 8 * hi][col] = v > 0.f ? v : 0.f;
        }
    }
    __syncthreads();

    // ---- relayout LDS (row-major) back into A-fragments ----
    #pragma unroll
    for (int s = 0; s < 32; ++s) {
        a[s].x = t_s[m][4 * s + 2 * hi];
        a[s].y = t_s[m][4 * s + 2 * hi + 1];
    }

    // ---- GEMM2: o = t @ W2 + b2 ----
    #pragma unroll
    for (int nt = 0; nt < 8; ++nt) {
        const int col = nt * 16 + m;
        v8f c = {};
        #pragma unroll
        for (int s = 0; s < 32; ++s) {
            const int kb = 4 * s + 2 * hi;
            v2f b;
            b.x = W2[(size_t)kb * DIM + col];
            b.y = W2[(size_t)(kb + 1) * DIM + col];
            c = __builtin_amdgcn_wmma_f32_16x16x4_f32(
                    false, a[s], false, b, (short)0, c, false, false);
        }
        const float bias = b2[col];
        #pragma unroll
        for (int r = 0; r < 8; ++r) {
            o_s[r + 8 * hi][col] = c[r] + bias;
        }
    }
    __syncthreads();

    // ---- LayerNorm stats: lanes 0..15 each reduce one row of 128 ----
    if (lane < 16) {
        float sum = 0.f, sq = 0.f;
        #pragma unroll 4
        for (int k = 0; k < DIM; ++k) {
            float v = o_s[lane][k];
            sum += v;
            sq  += v * v;
        }
        float mu  = sum * (1.0f / DIM);
        float var = sq * (1.0f / DIM) - mu * mu;
        mu_s[lane] = mu;
        rs_s[lane] = rsqrtf(var + 1e-5f);
    }
    __syncthreads();

    // ---- normalize + affine, coalesced 512B stores per row ----
    #pragma unroll
    for (int row = 0; row < 16; ++row) {
        const float muv = mu_s[row];
        const float rsv = rs_s[row];
        #pragma unroll
        for (int q = 0; q < 4; ++q) {
            int col = lane * 4 + q;
            out[(size_t)(base + row) * DIM + col] =
                (o_s[row][col] - muv) * rsv * gamma[col] + beta[col];
        }
    }
}

// ---------------------------------------------------------------------------
extern "C" void kernel_launch(void* const* d_in, const int* in_sizes, int n_in,
                              void* d_out, int out_size, void* d_ws, size_t ws_size,
                              hipStream_t stream) {
    (void)in_sizes; (void)n_in; (void)out_size; (void)ws_size;

    const float* x     = (const float*)d_in[0];
    const int*   ei    = (const int*)  d_in[1];
    const float* ea    = (const float*)d_in[2];
    const float* We    = (const float*)d_in[3];
    const float* be    = (const float*)d_in[4];
    const float* W1    = (const float*)d_in[5];
    const float* b1    = (const float*)d_in[6];
    const float* W2    = (const float*)d_in[7];
    const float* b2    = (const float*)d_in[8];
    const float* eps   = (const float*)d_in[9];
    const float* gamma = (const float*)d_in[10];
    const float* beta  = (const float*)d_in[11];
    float* out = (float*)d_out;
    float* agg = (float*)d_ws;                     // [N, 128] fp32 scratch

    const int n4 = N_NODES * DIM / 4;              // 3.2M float4
    zero_agg_kernel<<<(n4 + 255) / 256, 256, 0, stream>>>((float4*)agg, n4);

    // 600000/16 = 37500 edge tiles, 4 waves per 128-thread block
    edge_kernel<<<N_EDGES / 16 / 4, 128, 0, stream>>>(x, ei, ea, We, be, agg);

    // 100000/16 = 6250 node tiles, one wave32 per block
    node_kernel<<<N_NODES / 16, 32, 0, stream>>>(x, agg, W1, b1, W2, b2,
                                                 eps, gamma, beta, out);
}